// GraphAutoencoder_23390391894414
// MI455X (gfx1250) — compile-verified
//
#include <hip/hip_runtime.h>
#include <hip/hip_bf16.h>

typedef __attribute__((ext_vector_type(2))) float v2f;
typedef __attribute__((ext_vector_type(8))) float v8f;

#define NFEAT 128

// ---------------------------------------------------------------------------
// Degree / normalization kernels
// ---------------------------------------------------------------------------
__global__ void init_deg(float* __restrict__ deg, int n) {
    int i = blockIdx.x * blockDim.x + threadIdx.x;
    if (i < n) deg[i] = 1.0f;  // self-loop contributes 1 to in-degree
}

__global__ void deg_edges(const int* __restrict__ ecol, float* __restrict__ deg, int nE) {
    int e = blockIdx.x * blockDim.x + threadIdx.x;
    if (e < nE) atomicAdd(&deg[ecol[e]], 1.0f);
}

__global__ void deg_to_dis(float* __restrict__ deg, int n) {
    int i = blockIdx.x * blockDim.x + threadIdx.x;
    if (i < n) {
        float d = deg[i];
        deg[i] = d > 0.0f ? rsqrtf(d) : 0.0f;
    }
}

// ---------------------------------------------------------------------------
// fp32 WMMA GEMM:  out[m,n] = epilogue( sum_k A[m,k] * W[k,n] )
//   EPI 0: out = dis[m] * acc          (GCN "scaled" epilogue)
//   EPI 1: out = relu(acc + bias[n])   (decoder hidden)
//   EPI 2: out = acc + bias[n]         (decoder output)
// Block = 256 threads = 8 wave32; block tile = 128 rows x NOUT cols.
// Per ISA layouts (wave32, V_WMMA_F32_16X16X4_F32):
//   A 16x4 : lane half h, VGPR j -> A[m][2h+j]
//   B 4x16 : lane half h, VGPR j -> B[2h+j][n]
//   C 16x16: VGPR j -> row 8h+j, col = lane&15
// W is staged in LDS pair-interleaved: rows (2p, 2p+1) interleaved per column
// so one ds_load_b64 yields a full B fragment. Pair-row stride % 64 == 32
// words so the two half-waves (pair p vs p+1) hit disjoint LDS bank halves.
// ---------------------------------------------------------------------------
template <int K, int NOUT, int EPI>
__global__ __launch_bounds__(256) void gemm_wmma(
    const float* __restrict__ A, const float* __restrict__ W,
    const float* __restrict__ bias, const float* __restrict__ dis,
    float* __restrict__ out, int nRows) {
    constexpr int KC    = 64;              // K-chunk staged in LDS
    constexpr int PAIRS = KC / 2;          // interleaved row pairs per chunk
    constexpr int LDWP  = 2 * NOUT + 32;   // words per pair-row; %64 == 32
    constexpr int NT    = NOUT / 16;       // 16-wide N tiles per wave
    __shared__ float wlds[PAIRS * LDWP];

    const int lane  = threadIdx.x & 31;
    const int wave  = threadIdx.x >> 5;
    const int h     = lane >> 4;   // half-wave (K split)
    const int ln    = lane & 15;   // N / M index within tile
    const int mBase = blockIdx.x * 128 + wave * 16;
    const int m     = mBase + ln;
    const int mc    = m < nRows ? m : (nRows - 1);
    const bool mval = (m < nRows);
    const float* arow = A + (size_t)mc * K;

    v8f acc[NT] = {};

    for (int kc = 0; kc < K; kc += KC) {
        __syncthreads();
        // cooperative stage of W rows [kc, kc+KC), pair-interleaved
        for (int idx = threadIdx.x; idx < KC * NOUT; idx += 256) {
            int k = idx / NOUT;           // coalesced global reads over n
            int n = idx - k * NOUT;
            wlds[(k >> 1) * LDWP + 2 * n + (k & 1)] = W[(size_t)(kc + k) * NOUT + n];
        }
        __syncthreads();

        for (int k0 = 0; k0 < KC; k0 += 4) {
            v2f av = *(const v2f*)(arow + kc + k0 + 2 * h);
            v2f a;
            a.x = mval ? av.x : 0.0f;   // select, not branch: EXEC stays all-1s
            a.y = mval ? av.y : 0.0f;

            // hoist all NT B fragments: batch ds_load_b64, single wait,
            // then back-to-back WMMAs
            const float* brow = wlds + ((k0 >> 1) + h) * LDWP;
            v2f b[NT];
#pragma unroll
            for (int t = 0; t < NT; ++t)
                b[t] = *(const v2f*)(brow + 2 * (t * 16 + ln));
#pragma unroll
            for (int t = 0; t < NT; ++t)
                acc[t] = __builtin_amdgcn_wmma_f32_16x16x4_f32(
                    false, a, false, b[t], (short)0, acc[t], false, false);
        }
    }

    // epilogue + store (divergence only after all WMMAs are done)
#pragma unroll
    for (int t = 0; t < NT; ++t) {
#pragma unroll
        for (int j = 0; j < 8; ++j) {
            int rm = mBase + 8 * h + j;
            if (rm < nRows) {
                int cn  = t * 16 + ln;
                float v = acc[t][j];
                if constexpr (EPI == 0) {
                    v *= dis[rm];
                } else if constexpr (EPI == 1) {
                    v += bias[cn];
                    v = v > 0.0f ? v : 0.0f;
                } else {
                    v += bias[cn];
                }
                out[(size_t)rm * NOUT + cn] = v;
            }
        }
    }
}

// ---------------------------------------------------------------------------
// Edge scatter-add: agg[col] += scaled[row], F floats per edge, float4 lanes.
// ---------------------------------------------------------------------------
template <int F>
__global__ __launch_bounds__(256) void edge_scatter(
    const int* __restrict__ erow, const int* __restrict__ ecol,
    const float* __restrict__ scaled, float* __restrict__ agg, int nEdges) {
    constexpr int LPE = F / 4;        // lanes per edge
    constexpr int EPB = 256 / LPE;    // edges per block
    int e = blockIdx.x * EPB + threadIdx.x / LPE;
    if (e >= nEdges) return;
    int j = (threadIdx.x % LPE) * 4;
    int r = erow[e];
    int c = ecol[e];
    const float4 v = *(const float4*)(scaled + (size_t)r * F + j);
    float* dst = agg + (size_t)c * F + j;
    atomicAdd(dst + 0, v.x);
    atomicAdd(dst + 1, v.y);
    atomicAdd(dst + 2, v.z);
    atomicAdd(dst + 3, v.w);
}

// ---------------------------------------------------------------------------
// z[i,f] = relu( dis[i] * (agg[i,f] + scaled[i,f]) + bias[f] )
// (the "+scaled[i,f]" term is the self-loop: dis^2 * xW)
// ---------------------------------------------------------------------------
template <int F>
__global__ __launch_bounds__(256) void finalize_gcn(
    const float* __restrict__ agg, const float* __restrict__ scaled,
    const float* __restrict__ dis, const float* __restrict__ bias,
    float* __restrict__ z, int n) {
    constexpr int CPE = F / 4;  // float4 chunks per node
    int t = blockIdx.x * blockDim.x + threadIdx.x;
    if (t >= n * CPE) return;
    int i = t / CPE;
    int j = (t - i * CPE) * 4;
    float d = dis[i];
    const float4 a4 = *(const float4*)(agg + (size_t)i * F + j);
    const float4 s4 = *(const float4*)(scaled + (size_t)i * F + j);
    const float4 b4 = *(const float4*)(bias + j);
    float4 o;
    o.x = fmaxf(d * (a4.x + s4.x) + b4.x, 0.0f);
    o.y = fmaxf(d * (a4.y + s4.y) + b4.y, 0.0f);
    o.z = fmaxf(d * (a4.z + s4.z) + b4.z, 0.0f);
    o.w = fmaxf(d * (a4.w + s4.w) + b4.w, 0.0f);
    *(float4*)(z + (size_t)i * F + j) = o;
}

// ---------------------------------------------------------------------------
extern "C" void kernel_launch(void* const* d_in, const int* in_sizes, int n_in,
                              void* d_out, int out_size, void* d_ws, size_t ws_size,
                              hipStream_t stream) {
    const float* x     = (const float*)d_in[0];
    const int*   eidx  = (const int*)d_in[1];
    const float* Wenc1 = (const float*)d_in[2];
    const float* benc1 = (const float*)d_in[3];
    const float* Wenc2 = (const float*)d_in[4];
    const float* benc2 = (const float*)d_in[5];
    const float* Wdec1 = (const float*)d_in[6];
    const float* bdec1 = (const float*)d_in[7];
    const float* Wdec2 = (const float*)d_in[8];
    const float* bdec2 = (const float*)d_in[9];

    const int N = in_sizes[0] / NFEAT;
    const int E = in_sizes[1] / 2;
    const int* erow = eidx;       // edge_index[0] = source
    const int* ecol = eidx + E;   // edge_index[1] = target

    // workspace carve-up
    char* ws = (char*)d_ws;
    size_t off = 0;
    auto carve = [&](size_t bytes) {
        void* p = ws + off;
        off = (off + bytes + 255) & ~(size_t)255;
        return p;
    };
    float* dis  = (float*)carve((size_t)N * 4);
    float* bufA = (float*)carve((size_t)N * 128 * 4);  // scaled / z3
    float* bufB = (float*)carve((size_t)N * 128 * 4);  // agg
    float* bufC = (float*)carve((size_t)N * 128 * 4);  // z1 / z2

    dim3 blk(256);
    int nbN = (N + 255) / 256;
    int nbG = (N + 127) / 128;

    // symmetric deg^{-1/2}
    init_deg<<<nbN, blk, 0, stream>>>(dis, N);
    deg_edges<<<(E + 255) / 256, blk, 0, stream>>>(ecol, dis, E);
    deg_to_dis<<<nbN, blk, 0, stream>>>(dis, N);

    // ---- encoder GCN layer 1 (128 -> 128) ----
    gemm_wmma<128, 128, 0><<<nbG, blk, 0, stream>>>(x, Wenc1, nullptr, dis, bufA, N);
    hipMemsetAsync(bufB, 0, (size_t)N * 128 * 4, stream);
    edge_scatter<128><<<(E + 7) / 8, blk, 0, stream>>>(erow, ecol, bufA, bufB, E);
    finalize_gcn<128><<<((size_t)N * 32 + 255) / 256, blk, 0, stream>>>(bufB, bufA, dis, benc1, bufC, N);

    // ---- encoder GCN layer 2 (128 -> 64) ----
    gemm_wmma<128, 64, 0><<<nbG, blk, 0, stream>>>(bufC, Wenc2, nullptr, dis, bufA, N);
    hipMemsetAsync(bufB, 0, (size_t)N * 64 * 4, stream);
    edge_scatter<64><<<(E + 15) / 16, blk, 0, stream>>>(erow, ecol, bufA, bufB, E);
    finalize_gcn<64><<<((size_t)N * 16 + 255) / 256, blk, 0, stream>>>(bufB, bufA, dis, benc2, bufC, N);

    // ---- decoder linear 1 (64 -> 128) + relu ----
    gemm_wmma<64, 128, 1><<<nbG, blk, 0, stream>>>(bufC, Wdec1, bdec1, dis, bufA, N);

    // ---- decoder linear 2 (128 -> 128), no relu, straight to d_out ----
    gemm_wmma<128, 128, 2><<<nbG, blk, 0, stream>>>(bufA, Wdec2, bdec2, dis, (float*)d_out, N);
}